// GaussianMomentDescriptor_24953759990492
// MI455X (gfx1250) — compile-verified
//
#include <hip/hip_runtime.h>
#include <hip/hip_bf16.h>

#define N_RADIAL   5
#define N_BASIS    7
#define N_SPECIES  119
#define R_MAX      6.0f
#define R_MIN      0.5f

// compact moment layout per atom (100 floats):
//  [0..4]   m0 (5)
//  [5..19]  m1 (5 x 3)
//  [20..49] m2 unique (5 x 6), pairs (i<=j) in order 00,01,02,11,12,22
//  [50..99] m3 unique (5 x 10), triples (i<=j<=k) lexicographic
#define CM_STRIDE 100

// expanded per-atom LDS layout (200 floats):
//  [0..4] m0, [5..19] m1, [20..64] m2 full (r*9+i*3+j), [65..199] m3 full (r*27+i*9+j*3+k)
#define LDS_STRIDE 200

typedef __attribute__((ext_vector_type(2))) float v2f;
typedef __attribute__((ext_vector_type(8))) float v8f;

// ---------------------------------------------------------------- zero ws
__global__ void gm_zero_kernel(float* __restrict__ p, int n) {
    int i = blockIdx.x * blockDim.x + threadIdx.x;
    if (i < n) p[i] = 0.0f;
}

// ---------------------------------------------------------------- edges
__global__ __launch_bounds__(256)
void gm_edge_kernel(const float* __restrict__ drv,
                    const int*   __restrict__ Z,
                    const int*   __restrict__ nbr,
                    const float* __restrict__ emb,
                    float*       __restrict__ m,
                    int nnbr) {
    int n = blockIdx.x * blockDim.x + threadIdx.x;
    if (n >= nnbr) return;
    int ii = nbr[n];
    int jj = nbr[nnbr + n];
    if (ii == jj) return;                       // radial masked to zero in reference

    float x = drv[3 * n + 0];
    float y = drv[3 * n + 1];
    float z = drv[3 * n + 2];
    float dr  = sqrtf(x * x + y * y + z * z);
    float inv = 1.0f / (dr + 1e-5f);
    float d0 = x * inv, d1 = y * inv, d2 = z * inv;

    // basis: rad_norm * exp(-betta*(dr-shift_b)^2), betta=49/36, rad_norm=(2b/pi)^0.25
    const float betta    = 1.36111111f;
    const float rad_norm = 0.96482050f;
    float basis[N_BASIS];
#pragma unroll
    for (int b = 0; b < N_BASIS; ++b) {
        float t = dr - (R_MIN + ((R_MAX - R_MIN) / (float)N_BASIS) * (float)b);
        basis[b] = rad_norm * __expf(-betta * t * t);
    }
    float cutoff = 0.5f * (__cosf(0.52359878f * fminf(dr, R_MAX)) + 1.0f);
    const float embed_norm = 0.37796447f;       // 1/sqrt(7)
    float cw = cutoff * embed_norm;

    const float* cf = emb + (size_t)(Z[ii] * N_SPECIES + Z[jj]) * (N_RADIAL * N_BASIS);
    float radial[N_RADIAL];
#pragma unroll
    for (int r = 0; r < N_RADIAL; ++r) {
        float a = 0.0f;
#pragma unroll
        for (int b = 0; b < N_BASIS; ++b) a += cf[r * N_BASIS + b] * basis[b];
        radial[r] = a * cw;
    }

    // symmetric-unique direction products
    float p2[6];
    p2[0] = d0 * d0; p2[1] = d0 * d1; p2[2] = d0 * d2;
    p2[3] = d1 * d1; p2[4] = d1 * d2; p2[5] = d2 * d2;
    float p3[10];
    p3[0] = p2[0] * d0; p3[1] = p2[0] * d1; p3[2] = p2[0] * d2;
    p3[3] = p2[1] * d1; p3[4] = p2[1] * d2; p3[5] = p2[2] * d2;
    p3[6] = p2[3] * d1; p3[7] = p2[3] * d2; p3[8] = p2[4] * d2;
    p3[9] = p2[5] * d2;

    float* base = m + (size_t)jj * CM_STRIDE;
#pragma unroll
    for (int r = 0; r < N_RADIAL; ++r) {
        float rad = radial[r];
        atomicAdd(base + r, rad);
        atomicAdd(base + 5  + r * 3 + 0, rad * d0);
        atomicAdd(base + 5  + r * 3 + 1, rad * d1);
        atomicAdd(base + 5  + r * 3 + 2, rad * d2);
#pragma unroll
        for (int q = 0; q < 6; ++q)  atomicAdd(base + 20 + r * 6  + q, rad * p2[q]);
#pragma unroll
        for (int q = 0; q < 10; ++q) atomicAdd(base + 50 + r * 10 + q, rad * p3[q]);
    }
}

// ---------------------------------------------------------------- helpers
__device__ __forceinline__ int pair2_idx(int a, int b) {   // (a<=b), n=3
    return a * 3 - (a * (a + 1)) / 2 + b;                  // 00,01,02,11,12,22
}
__device__ __forceinline__ void pair_from_p(int p, int& r, int& s) {  // n=5 triu
    int r0 = 0;
    while (p >= (5 - r0)) { p -= (5 - r0); ++r0; }
    r = r0; s = r0 + p;
}
__device__ __forceinline__ void trip_from_t(int t, int& a, int& b, int& c) { // n=5, a<=b<=c
    a = 0;
    while (t >= (5 - a) * (6 - a) / 2) { t -= (5 - a) * (6 - a) / 2; ++a; }
    b = a;
    while (t >= (5 - b)) { t -= (5 - b); ++b; }
    c = b + t;
}
__device__ __forceinline__ int pidx5(int r, int s) {       // triu(5) flat index, r<=s
    return r * 5 - (r * (r - 1)) / 2 + (s - r);
}

// Gram S = M * M^T (M is 5 x K, contiguous in LDS), via V_WMMA_F32_16X16X4_F32.
// Writes the 15 triu entries to outp[pidx5(r,s)].
template <int K>
__device__ __forceinline__ void gram_wmma(const float* __restrict__ Mb,
                                          float* __restrict__ outp, int lane) {
    v8f acc = {};
    int row = lane & 15;
    int kh  = (lane < 16) ? 0 : 2;
#pragma unroll
    for (int k0 = 0; k0 < K; k0 += 4) {
        int ka = k0 + kh;
        v2f av; av.x = 0.0f; av.y = 0.0f;
        if (row < 5) {
            if (ka     < K) av.x = Mb[row * K + ka];
            if (ka + 1 < K) av.y = Mb[row * K + ka + 1];
        }
        // A(16x4) == B(4x16)^T fragment for a Gram product (same per-lane data).
        acc = __builtin_amdgcn_wmma_f32_16x16x4_f32(false, av, false, av,
                                                    (short)0, acc, false, false);
    }
    if (lane < 5) {
        int s = lane;                           // column index (N) for lanes 0..15
        float col[8] = {acc[0], acc[1], acc[2], acc[3], acc[4], acc[5], acc[6], acc[7]};
#pragma unroll
        for (int r = 0; r < 5; ++r)
            if (r <= s) outp[pidx5(r, s)] = col[r];
    }
}

// ---------------------------------------------------------------- per-atom contraction
__global__ __launch_bounds__(256)
void gm_contract_kernel(const float* __restrict__ m,
                        float*       __restrict__ out,
                        int natoms) {
    __shared__ float lds[8 * LDS_STRIDE];
    int wave = threadIdx.x >> 5;
    int lane = threadIdx.x & 31;
    int a    = blockIdx.x * 8 + wave;
    bool active = (a < natoms);

    float* la = &lds[wave * LDS_STRIDE];
    if (active) {
        const float* cm = m + (size_t)a * CM_STRIDE;
        // m0 + m1 are already dense
        for (int f = lane; f < 20; f += 32) la[f] = cm[f];
        // expand m2 (sym) -> full 5x3x3
        for (int f = lane; f < 45; f += 32) {
            int r = f / 9, ij = f % 9, i = ij / 3, j = ij % 3;
            int aa = min(i, j), bb = max(i, j);
            la[20 + f] = cm[20 + r * 6 + pair2_idx(aa, bb)];
        }
        // expand m3 (sym) -> full 5x3x3x3
        for (int f = lane; f < 135; f += 32) {
            int r = f / 27, rem = f % 27;
            int i = rem / 9, j = (rem / 3) % 3, k = rem % 3;
            int aa = min(i, min(j, k));
            int cc = max(i, max(j, k));
            int bb = i + j + k - aa - cc;
            int base3 = (aa == 0) ? 0 : ((aa == 1) ? 6 : 9);
            int sub   = pair2_idx(bb, cc) - pair2_idx(aa, aa);
            la[65 + f] = cm[50 + r * 10 + base3 + sub];
        }
    }
    __syncthreads();
    if (!active) return;

    const float* m0  = la;
    const float* m1  = la + 5;
    const float* m2f = la + 20;
    const float* m3f = la + 65;
    float* op = out + (size_t)a * 360;

    // c0 (5)
    if (lane < 5) op[lane] = m0[lane];

    // c1/c2/c3 Gram matrices via WMMA (15 each)
    gram_wmma<3>(m1, op + 5, lane);
    gram_wmma<9>(m2f, op + 20, lane);
    gram_wmma<27>(m3f, op + 35, lane);

    // c4 (35): sum_ijk m2[r,i,j] m2[s,i,k] m2[u,j,k], (r<=s<=u)
    for (int t = lane; t < 35; t += 32) {
        int r, s, u; trip_from_t(t, r, s, u);
        const float* R = m2f + r * 9;
        const float* S = m2f + s * 9;
        const float* U = m2f + u * 9;
        float acc = 0.0f;
#pragma unroll
        for (int i = 0; i < 3; ++i)
#pragma unroll
            for (int j = 0; j < 3; ++j)
#pragma unroll
                for (int k = 0; k < 3; ++k)
                    acc += R[i * 3 + j] * S[i * 3 + k] * U[j * 3 + k];
        op[50 + t] = acc;
    }

    // c5 (75): sum_ij m1[r,i] m1[s,j] m2[t,i,j], pairs (r<=s) x t
    for (int q = lane; q < 75; q += 32) {
        int p = q / 5, t = q % 5, r, s; pair_from_p(p, r, s);
        float acc = 0.0f;
#pragma unroll
        for (int i = 0; i < 3; ++i)
#pragma unroll
            for (int j = 0; j < 3; ++j)
                acc += m1[r * 3 + i] * m1[s * 3 + j] * m2f[t * 9 + i * 3 + j];
        op[85 + q] = acc;
    }

    // c6 (75): sum_ijkl m3[r,i,j,k] m3[s,i,j,l] m2[t,k,l], pairs (r<=s) x t
    for (int q = lane; q < 75; q += 32) {
        int p = q / 5, t = q % 5, r, s; pair_from_p(p, r, s);
        float acc = 0.0f;
#pragma unroll
        for (int k = 0; k < 3; ++k)
#pragma unroll
            for (int l = 0; l < 3; ++l) {
                float w = m2f[t * 9 + k * 3 + l];
                float dot = 0.0f;
#pragma unroll
                for (int i = 0; i < 3; ++i)
#pragma unroll
                    for (int j = 0; j < 3; ++j)
                        dot += m3f[r * 27 + i * 9 + j * 3 + k] *
                               m3f[s * 27 + i * 9 + j * 3 + l];
                acc += dot * w;
            }
        op[160 + q] = acc;
    }

    // c7 (125): sum_ijk m3[r,i,j,k] m2[s,i,j] m1[t,k]  (full r,s,t)
    for (int q = lane; q < 125; q += 32) {
        int r = q / 25, s = (q / 5) % 5, t = q % 5;
        float acc = 0.0f;
#pragma unroll
        for (int i = 0; i < 3; ++i)
#pragma unroll
            for (int j = 0; j < 3; ++j) {
                float w = m2f[s * 9 + i * 3 + j];
#pragma unroll
                for (int k = 0; k < 3; ++k)
                    acc += m3f[r * 27 + i * 9 + j * 3 + k] * w * m1[t * 3 + k];
            }
        op[235 + q] = acc;
    }
}

// ---------------------------------------------------------------- launch
extern "C" void kernel_launch(void* const* d_in, const int* in_sizes, int n_in,
                              void* d_out, int out_size, void* d_ws, size_t ws_size,
                              hipStream_t stream) {
    const float* drv = (const float*)d_in[0];
    const int*   Z   = (const int*)d_in[1];
    const int*   nbr = (const int*)d_in[2];
    const float* emb = (const float*)d_in[3];
    float*       out = (float*)d_out;

    int nnbr   = in_sizes[0] / 3;
    int natoms = in_sizes[1];

    float* mws = (float*)d_ws;                  // natoms * 100 floats (8 MB)
    int nm = natoms * CM_STRIDE;

    gm_zero_kernel<<<(nm + 255) / 256, 256, 0, stream>>>(mws, nm);
    gm_edge_kernel<<<(nnbr + 255) / 256, 256, 0, stream>>>(drv, Z, nbr, emb, mws, nnbr);
    gm_contract_kernel<<<(natoms + 7) / 8, 256, 0, stream>>>(mws, out, natoms);
}